// FocalLoss_5669356831818
// MI455X (gfx1250) — compile-verified
//
#include <hip/hip_runtime.h>
#include <hip/hip_bf16.h>

#define NB    8      // batch
#define NA    65536  // anchors
#define NC    80     // classes
#define NM    32     // boxes per image
#define BLKX  128    // blocks per image
#define TPB   256    // threads per block
#define APB   (NA / BLKX)        // anchors per block = 512
#define ITER  (APB / TPB)        // anchors per thread = 2

typedef __attribute__((ext_vector_type(2))) float v2f;
typedef __attribute__((ext_vector_type(4))) float v4f;
typedef __attribute__((ext_vector_type(8))) float v8f;

// --- exact 32-lane f32 sum via V_WMMA_F32_16X16X4_F32 ------------------
// A[m][k]: lane's x in slot 0, 0 in slot 1; B = all ones.
// D[m][n] = rowsum_m for every n.  Sum of a lane's 8 D VGPRs gives
// sum over half the rows; xor-16 shuffle adds the other half.  This is
// layout-robust: every A entry contributes exactly once to each column.
__device__ __forceinline__ float wave_sum_wmma(float x) {
    v2f a; a[0] = x;   a[1] = 0.0f;
    v2f b; b[0] = 1.0f; b[1] = 1.0f;
    v8f c = {};
    c = __builtin_amdgcn_wmma_f32_16x16x4_f32(
            /*neg_a=*/false, a, /*neg_b=*/false, b,
            /*c_mod=*/(short)0, c, /*reuse_a=*/false, /*reuse_b=*/false);
    float t = c[0] + c[1] + c[2] + c[3] + c[4] + c[5] + c[6] + c[7];
    t += __shfl_xor(t, 16, 32);
    return t;
}

__device__ __forceinline__ float clampp(float p) {
    return fminf(fmaxf(p, 5e-4f), 0.9995f);
}
// (1-alpha) * p^2 * (-log(1-p))   with alpha = 0.25, gamma = 2
__device__ __forceinline__ float neg_term(float p) {
    p = clampp(p);
    return 0.75f * p * p * (-__logf(1.0f - p));
}
__device__ __forceinline__ float smooth_l1(float d) {
    d = fabsf(d);
    return (d <= (1.0f / 9.0f)) ? 4.5f * d * d : d - (0.5f / 9.0f);
}

// part[(img*BLKX + blk)*3 + {0,1,2}] = {cls_sum, reg_sum, pos_count}
__global__ void __launch_bounds__(TPB)
focal_main(const float* __restrict__ cls,
           const float* __restrict__ reg,
           const float* __restrict__ anc,
           const float* __restrict__ ann,
           float* __restrict__ part) {
    const int img = blockIdx.y;
    const int tid = threadIdx.x;

    __shared__ float bx1[NM], by1[NM], bx2[NM], by2[NM], bcl[NM], bar[NM];
    __shared__ float wsum[TPB / 32][3];

    if (tid < NM) {
        const float* bp = ann + ((size_t)img * NM + tid) * 5;
        float x1 = bp[0], y1 = bp[1], x2 = bp[2], y2 = bp[3];
        bx1[tid] = x1; by1[tid] = y1; bx2[tid] = x2; by2[tid] = y2;
        bcl[tid] = bp[4];
        bar[tid] = (x2 - x1) * (y2 - y1);
    }
    __syncthreads();

    float s_cls = 0.0f, s_reg = 0.0f, s_pos = 0.0f;
    const int a0 = blockIdx.x * APB;

    for (int it = 0; it < ITER; ++it) {
        const int a = a0 + it * TPB + tid;

        // anchor row: [y1, x1, y2, x2]
        v4f av = *(const v4f*)(anc + 4 * (size_t)a);
        const float ay1 = av.x, ax1 = av.y, ay2 = av.z, ax2 = av.w;
        const float aw = ax2 - ax1, ah = ay2 - ay1;
        const float aarea = aw * ah;

        // IoU-max over 32 boxes (scale-invariant: skip /s normalization)
        float best = -1.0f; int bestj = 0;
        #pragma unroll 8
        for (int j = 0; j < NM; ++j) {
            float iw = fminf(ax2, bx2[j]) - fmaxf(ax1, bx1[j]);
            float ih = fminf(ay2, by2[j]) - fmaxf(ay1, by1[j]);
            iw = fmaxf(iw, 0.0f); ih = fmaxf(ih, 0.0f);
            float inter = iw * ih;
            float ua = fmaxf(aarea + bar[j] - inter, 1e-8f);
            float iou = inter / ua;
            if (iou > best) { best = iou; bestj = j; }
        }
        const bool pos = best >= 0.5f;
        const bool ign = (best >= 0.4f) && !pos;

        // classification loss: skip ignored anchors (saves bandwidth too)
        if (!ign) {
            const float* crow = cls + ((size_t)img * NA + (size_t)a) * NC;
            const v4f* cp = (const v4f*)crow;
            float base = 0.0f;
            #pragma unroll 5
            for (int q = 0; q < NC / 4; ++q) {
                v4f p4 = __builtin_nontemporal_load(&cp[q]);   // streamed once
                base += neg_term(p4.x) + neg_term(p4.y)
                      + neg_term(p4.z) + neg_term(p4.w);
            }
            if (pos) {  // swap class-k term from neg to pos form
                float pk = clampp(crow[(int)bcl[bestj]]);
                float omp = 1.0f - pk;
                base += 0.25f * omp * omp * (-__logf(pk))
                      - 0.75f * pk * pk * (-__logf(omp));
            }
            s_cls += base;
        }

        // regression loss (positives only)
        if (pos) {
            s_pos += 1.0f;
            float gx1 = bx1[bestj], gy1 = by1[bestj];
            float gx2 = bx2[bestj], gy2 = by2[bestj];
            float gw = gx2 - gx1, gh = gy2 - gy1;
            float gcx = gx1 + 0.5f * gw, gcy = gy1 + 0.5f * gh;
            gw = fmaxf(gw, 1.0f); gh = fmaxf(gh, 1.0f);
            float acx = ax1 + 0.5f * aw, acy = ay1 + 0.5f * ah;
            float t0 = (gcy - acy) / ah;
            float t1 = (gcx - acx) / aw;
            float t2 = __logf(gh / ah);
            float t3 = __logf(gw / aw);
            v4f rg = *(const v4f*)(reg + ((size_t)img * NA + (size_t)a) * 4);
            s_reg += smooth_l1(t0 - rg.x) + smooth_l1(t1 - rg.y)
                   + smooth_l1(t2 - rg.z) + smooth_l1(t3 - rg.w);
        }
    }

    // block tail: EXEC is all-ones here (required for WMMA)
    float tc = wave_sum_wmma(s_cls);
    float tr = wave_sum_wmma(s_reg);
    float tp = wave_sum_wmma(s_pos);

    const int wave = tid >> 5, lane = tid & 31;
    if (lane == 0) { wsum[wave][0] = tc; wsum[wave][1] = tr; wsum[wave][2] = tp; }
    __syncthreads();
    if (tid == 0) {
        float a0s = 0.0f, a1s = 0.0f, a2s = 0.0f;
        #pragma unroll
        for (int w = 0; w < TPB / 32; ++w) {
            a0s += wsum[w][0]; a1s += wsum[w][1]; a2s += wsum[w][2];
        }
        float* p = part + ((size_t)img * BLKX + blockIdx.x) * 3;
        p[0] = a0s; p[1] = a1s; p[2] = a2s;  // full overwrite: no zeroing needed
    }
}

__global__ void focal_final(const float* __restrict__ part, float* __restrict__ out) {
    __shared__ float cimg[NB], rimg[NB];
    const int t = threadIdx.x;
    if (t < NB) {
        float sc = 0.0f, sr = 0.0f, sp = 0.0f;
        for (int blk = 0; blk < BLKX; ++blk) {
            const float* p = part + ((size_t)t * BLKX + blk) * 3;
            sc += p[0]; sr += p[1]; sp += p[2];
        }
        cimg[t] = sc / fmaxf(sp, 1.0f);
        rimg[t] = (sp > 0.0f) ? sr / fmaxf(sp * 4.0f, 1.0f) : 0.0f;
    }
    __syncthreads();
    if (t == 0) {
        float cm = 0.0f, rm = 0.0f;
        #pragma unroll
        for (int b = 0; b < NB; ++b) { cm += cimg[b]; rm += rimg[b]; }
        cm *= (1.0f / NB); rm *= (1.0f / NB);
        out[0] = cm + rm; out[1] = cm; out[2] = rm;
    }
}

extern "C" void kernel_launch(void* const* d_in, const int* in_sizes, int n_in,
                              void* d_out, int out_size, void* d_ws, size_t ws_size,
                              hipStream_t stream) {
    (void)in_sizes; (void)n_in; (void)out_size; (void)ws_size;
    const float* cls = (const float*)d_in[0];
    const float* reg = (const float*)d_in[1];
    const float* anc = (const float*)d_in[2];
    const float* ann = (const float*)d_in[3];
    float* part = (float*)d_ws;   // NB*BLKX*3 floats = 12 KB
    float* out  = (float*)d_out;

    focal_main <<<dim3(BLKX, NB), TPB, 0, stream>>>(cls, reg, anc, ann, part);
    focal_final<<<1, 32, 0, stream>>>(part, out);
}